// Actor_85968065397400
// MI455X (gfx1250) — compile-verified
//
#include <hip/hip_runtime.h>
#include <stdint.h>

// Greedy masked-argmax scan (B=65536, S=V=25), bandwidth-bound (~164 MB in, 13 MB out,
// ~7.6 us floor at 23.3 TB/s). CDNA5 plan:
//  - tensor_load_to_lds DMAs each block's contiguous 80,000-B slice into LDS in ONE
//    instruction (perfectly coalesced, no VGPR address traffic).
//  - single-wave workgroups: no barriers, just s_wait_tensorcnt.
//  - scan runs from LDS; per-thread stride 625 dwords (49 mod 64) -> bank-conflict-free.
//  - inner argmax is an adjacent-pair tournament (depth 5, ~12-way ILP) instead of a
//    25-deep serial chain: critical for latency hiding at 1 wave/SIMD.
//  - outputs staged in the (dead) tile LDS after the scan, pushed with coalesced
//    global_store_async_from_lds_b128.

#define TPB 32
#define SEQ 25
#define VOC 25
#define ROW_DW (SEQ * VOC)        // 625 dwords per batch row (2500 B, contiguous)
#define TILE_DW (TPB * ROW_DW)    // 20000 dwords = 80,000 B per workgroup tile
#define OUT_DW (TPB * SEQ)        // 800 dwords per output half per block (3200 B)

typedef unsigned int v4u __attribute__((ext_vector_type(4)));
typedef int          v8i __attribute__((ext_vector_type(8)));
typedef int          v4i __attribute__((ext_vector_type(4)));

__global__ __launch_bounds__(TPB) void greedy_argmax_scan(
    const float* __restrict__ matrix, float* __restrict__ out, int B)
{
    __shared__ float smem[TILE_DW];   // 80,000 B -> 4 single-wave WGs per WGP

    const int tid = threadIdx.x;
    const long long b0 = (long long)blockIdx.x * TPB;

    // ---------------- TDM: DMA 32 rows (contiguous 80,000 B) into LDS ----------------
    const unsigned long long gaddr =
        (unsigned long long)(const void*)(matrix + (size_t)b0 * ROW_DW);
    const unsigned lds_base = (unsigned)(unsigned long long)(const void*)&smem[0];

    v4u g0;
    g0[0] = 1u;                                   // count=1 (valid user descriptor)
    g0[1] = lds_base;                             // lds_addr (bytes)
    g0[2] = (unsigned)gaddr;                      // global_addr[31:0]
    g0[3] = (unsigned)((gaddr >> 32) & 0x01FFFFFFull) // global_addr[56:32]
          | 0x80000000u;                          // type=2 ("image") in bits[127:126]

    v8i g1 = {0, 0, 0, 0, 0, 0, 0, 0};
    g1[0] = 0x00020000;                           // data_size=2 (4 B); no multicast/pad
    g1[1] = (int)((TILE_DW & 0xFFFF) << 16);      // tensor_dim0[15:0]
    g1[2] = (int)(((TILE_DW >> 16) & 0xFFFF)      // tensor_dim0[31:16]
          | (1u << 16));                          // tensor_dim1 = 1
    g1[3] = (int)((TILE_DW & 0xFFFF) << 16);      // tile_dim0 = 20000 (1-D tile)
    g1[5] = TILE_DW;                              // tensor_dim0_stride[31:0]

    v4i g2 = {0, 0, 0, 0};
    v4i g3 = {0, 0, 0, 0};
    v8i g4 = {0, 0, 0, 0, 0, 0, 0, 0};            // 6-arg toolchain: extra group, zeroed

    // Issued once per wave (EXEC ignored by TDM); one wave per block.
    __builtin_amdgcn_tensor_load_to_lds(g0, g1, g2, g3, g4, 0);
    __builtin_amdgcn_s_wait_tensorcnt(0);
    asm volatile("" ::: "memory");                // compiler fence: LDS now valid

    // ---------------- per-thread greedy scan from LDS ----------------
    const float* row = &smem[tid * ROW_DW];       // stride 625 dw: conflict-free
    float acts[SEQ];
    float vals[SEQ];
    unsigned used = 0u;

#pragma unroll
    for (int s = 0; s < SEQ; ++s) {
        const float* sc = row + s * VOC;

        // Masked candidates (jnp.finfo(float32).min masking).
        float tv[VOC];
        int   ti[VOC];
#pragma unroll
        for (int v = 0; v < VOC; ++v) {
            float x = sc[v];
            tv[v] = ((used >> v) & 1u) ? -__FLT_MAX__ : x;
            ti[v] = v;
        }

        // Adjacent-pair tournament: strict '>' keeps the LEFT entry on ties, and
        // adjacent pairing means the left entry always covers smaller original
        // indices -> final winner is the first (lowest-index) maximum, matching
        // jnp.argmax. Depth: ceil(log2(25)) = 5 levels, merges independent per level.
#pragma unroll
        for (int w = VOC; w > 1; w = (w + 1) >> 1) {
#pragma unroll
            for (int k = 0; k + k + 1 < w; ++k) {
                float av = tv[2 * k], bv = tv[2 * k + 1];
                int   ai = ti[2 * k], bi = ti[2 * k + 1];
                bool take_b = bv > av;
                tv[k] = take_b ? bv : av;
                ti[k] = take_b ? bi : ai;
            }
            if (w & 1) {              // odd leftover passes through
                tv[w >> 1] = tv[w - 1];
                ti[w >> 1] = ti[w - 1];
            }
        }

        const int   besti = ti[0];
        const float bestv = tv[0];    // winner is never masked (S <= V)

        used |= (1u << besti);
        acts[s] = (float)besti;
        vals[s] = bestv;
    }

    // Wave executes in program order: all tile reads retired before the stores
    // below; fence stops the compiler from sinking stores into the scan.
    asm volatile("" ::: "memory");

    // ---------------- stage outputs in (dead) tile LDS, coalesced async store ----------------
    float* outA = &smem[0];
    float* outS = &smem[OUT_DW];
#pragma unroll
    for (int s = 0; s < SEQ; ++s) {
        outA[tid * SEQ + s] = acts[s];
        outS[tid * SEQ + s] = vals[s];
    }
    asm volatile("s_wait_dscnt 0x0" ::: "memory");   // LDS writes landed

    const unsigned ldsA = lds_base;
    const unsigned ldsS = lds_base + OUT_DW * 4;
    const unsigned long long dstA =
        (unsigned long long)(void*)(out + (size_t)b0 * SEQ);
    const unsigned long long dstS =
        (unsigned long long)(void*)(out + (size_t)B * SEQ + (size_t)b0 * SEQ);

    for (int c = tid * 16; c < OUT_DW * 4; c += TPB * 16) {   // 200 x 16B chunks
        asm volatile("global_store_async_from_lds_b128 %0, %1, off"
                     :: "v"(dstA + (unsigned long long)c), "v"(ldsA + (unsigned)c)
                     : "memory");
        asm volatile("global_store_async_from_lds_b128 %0, %1, off"
                     :: "v"(dstS + (unsigned long long)c), "v"(ldsS + (unsigned)c)
                     : "memory");
    }
    asm volatile("s_wait_asynccnt 0x0" ::: "memory");
}

extern "C" void kernel_launch(void* const* d_in, const int* in_sizes, int n_in,
                              void* d_out, int out_size, void* d_ws, size_t ws_size,
                              hipStream_t stream)
{
    (void)n_in; (void)out_size; (void)d_ws; (void)ws_size;
    const float* matrix = (const float*)d_in[0];
    float* out = (float*)d_out;
    const int B = in_sizes[0] / (SEQ * VOC);   // 65536
    const int blocks = B / TPB;                // 2048 (B is a multiple of 32)
    if (blocks > 0) {
        greedy_argmax_scan<<<blocks, TPB, 0, stream>>>(matrix, out, B);
    }
}